// ExpanderGCNNet_44744969290331
// MI455X (gfx1250) — compile-verified
//
#include <hip/hip_runtime.h>
#include <hip/hip_bf16.h>

#define N_NODES   50000
#define N_EDGES   600000
#define HID       128
#define N_LAYERS  4
#define N_GRAPHS  128
#define N_CLASSES 10
#define BN_EPS    1e-5f

typedef float v2f __attribute__((ext_vector_type(2)));
typedef float v8f __attribute__((ext_vector_type(8)));

#define LDSW_STRIDE 132   // 128 + 4 pad: even (8B-aligned b64 LDS reads), bank stride 4

__device__ __forceinline__ void atomAddF32(float* p, float v) {
    __hip_atomic_fetch_add(p, v, __ATOMIC_RELAXED, __HIP_MEMORY_SCOPE_AGENT);
}

// ---------------------------------------------------------------------------
// Vectorized zero fill (float4)
__global__ void zero_f32v4(float4* __restrict__ p, long n4) {
    long i = (long)blockIdx.x * blockDim.x + threadIdx.x;
    if (i < n4) {
        float4 z = {0.0f, 0.0f, 0.0f, 0.0f};
        p[i] = z;
    }
}

// ---------------------------------------------------------------------------
// Masked weights: wm[0:16384) = embed_W*embed_mask ; wm[16384:) = layer_W*layer_mask
__global__ void mask_weights(const float* __restrict__ eW, const float* __restrict__ eM,
                             const float* __restrict__ lW, const float* __restrict__ lM,
                             float* __restrict__ wm) {
    int i = blockIdx.x * blockDim.x + threadIdx.x;
    const int E = HID * HID;
    if (i >= E * (1 + N_LAYERS)) return;
    if (i < E) wm[i] = eW[i] * eM[i];
    else       wm[i] = lW[i - E] * lM[i - E];
}

// ---------------------------------------------------------------------------
// Z[N,128] = X[N,128] @ Wm[128,128]^T + bias   (fp32 WMMA 16x16x4)
// Weights staged via async Global->LDS DMA (ASYNCcnt path, no VGPR round-trip).
// mode 0: plain store (embedding)
// mode 1: z *= snorm[row]; store; atomically accumulate per-feature sum/sumsq
__global__ __launch_bounds__(256) void gemm_wmma(
    const float* __restrict__ X, const float* __restrict__ Wm,
    const float* __restrict__ bias, const float* __restrict__ snorm,
    float* __restrict__ Z, float* __restrict__ bn_sum, float* __restrict__ bn_sq,
    int N, int mode)
{
    __shared__ float ldsW[HID * LDSW_STRIDE];
    const int tid = threadIdx.x;

    // Stage masked weights (row-major Wm[n][k], padded row stride) into LDS
    // using GLOBAL_LOAD_ASYNC_TO_LDS_B128: one 16B transfer per lane per issue.
    // Low 32 bits of a generic pointer into the LDS aperture == LDS byte offset.
    for (int i = tid; i < HID * (HID / 4); i += 256) {
        int n  = i >> 5;            // 32 float4 per row
        int k4 = (i & 31) << 2;
        unsigned ldsoff = (unsigned)(uintptr_t)(&ldsW[n * LDSW_STRIDE + k4]);
        const float* g = Wm + (size_t)i * 4;
        asm volatile("global_load_async_to_lds_b128 %0, %1, off"
                     :: "v"(ldsoff), "v"(g)
                     : "memory");
    }
    asm volatile("s_wait_asynccnt 0x0" ::: "memory");  // own wave's DMA done
    __syncthreads();                                   // all waves' DMA visible

    const int wave = tid >> 5;
    const int lane = tid & 31;
    const int hi   = lane >> 4;          // half-wave select
    const int col  = lane & 15;          // N (and M for A-frag) within tile
    const int kb   = hi << 1;            // 0 or 2: K sub-offset per ISA layout
    const int m0   = blockIdx.x * 128 + wave * 16;

    int arow = m0 + col;                 // A-fragment row; clamp OOB (stores masked)
    if (arow >= N) arow = N - 1;
    const float* Xrow = X + (long)arow * HID + kb;

    v8f acc[8];
    #pragma unroll
    for (int nt = 0; nt < 8; ++nt) {
        float b = bias[nt * 16 + col];
        #pragma unroll
        for (int r = 0; r < 8; ++r) acc[nt][r] = b;   // bias fused into C
    }

    #pragma unroll 4
    for (int ks = 0; ks < HID; ks += 4) {
        // A 16x4 fp32 fragment: lanes0-15 K={ks,ks+1}, lanes16-31 K={ks+2,ks+3}
        v2f a = *(const v2f*)(Xrow + ks);
        #pragma unroll
        for (int nt = 0; nt < 8; ++nt) {
            // B 4x16 fragment (B = Wm^T): VGPR i = row K=kb+i, lane = N
            v2f b = *(const v2f*)(&ldsW[(nt * 16 + col) * LDSW_STRIDE + ks + kb]);
            acc[nt] = __builtin_amdgcn_wmma_f32_16x16x4_f32(
                false, a, false, b, (short)0, acc[nt], false, false);
        }
    }

    // D layout: VGPR r holds row M = r + 8*hi, lane = col
    if (mode == 0) {
        #pragma unroll
        for (int nt = 0; nt < 8; ++nt) {
            #pragma unroll
            for (int r = 0; r < 8; ++r) {
                int row = m0 + r + (hi << 3);
                if (row < N) Z[(long)row * HID + nt * 16 + col] = acc[nt][r];
            }
        }
    } else {
        float sn[8]; bool ok[8];
        #pragma unroll
        for (int r = 0; r < 8; ++r) {
            int row = m0 + r + (hi << 3);
            ok[r] = row < N;
            sn[r] = ok[r] ? snorm[row] : 0.0f;
        }
        #pragma unroll
        for (int nt = 0; nt < 8; ++nt) {
            float s = 0.0f, s2 = 0.0f;
            #pragma unroll
            for (int r = 0; r < 8; ++r) {
                int row = m0 + r + (hi << 3);
                float v = acc[nt][r] * sn[r];
                if (ok[r]) {
                    Z[(long)row * HID + nt * 16 + col] = v;
                    s += v; s2 += v * v;
                }
            }
            atomAddF32(&bn_sum[nt * 16 + col], s);
            atomAddF32(&bn_sq [nt * 16 + col], s2);
        }
    }
}

// ---------------------------------------------------------------------------
// Edge scatter: agg[dst] += h[src]. One wave per edge; lane = one float4 chunk.
__global__ __launch_bounds__(256) void edge_agg(
    const float* __restrict__ h, const int* __restrict__ src,
    const int* __restrict__ dst, float* __restrict__ agg, int nE)
{
    int e    = (blockIdx.x * 256 + threadIdx.x) >> 5;
    int lane = threadIdx.x & 31;
    if (e >= nE) return;
    int s = src[e], d = dst[e];                 // wave-uniform -> scalar loads
    float4 v = *(const float4*)(h + (long)s * HID + lane * 4);
    float* p = agg + (long)d * HID + lane * 4;
    atomAddF32(p + 0, v.x);
    atomAddF32(p + 1, v.y);
    atomAddF32(p + 2, v.z);
    atomAddF32(p + 3, v.w);
}

// ---------------------------------------------------------------------------
// h += relu( (z - mu)*rsqrt(var+eps)*gamma + beta )  — fused BN/act/residual
__global__ void bn_apply(const float* __restrict__ z, const float* __restrict__ bn_sum,
                         const float* __restrict__ bn_sq, const float* __restrict__ gamma,
                         const float* __restrict__ beta, float* __restrict__ h, int N)
{
    int idx = blockIdx.x * blockDim.x + threadIdx.x;     // one float4 per thread
    if (idx >= N * (HID / 4)) return;
    int col4 = (idx & 31) * 4;
    const float invN = 1.0f / (float)N;
    float4 zv = *(const float4*)(z + (long)idx * 4);
    float4 hv = *(const float4*)(h + (long)idx * 4);
    float zz[4] = {zv.x, zv.y, zv.z, zv.w};
    float hh[4] = {hv.x, hv.y, hv.z, hv.w};
    #pragma unroll
    for (int j = 0; j < 4; ++j) {
        int c = col4 + j;
        float mu  = bn_sum[c] * invN;
        float var = bn_sq[c] * invN - mu * mu;
        float t = (zz[j] - mu) * __frsqrt_rn(var + BN_EPS) * gamma[c] + beta[c];
        hh[j] += fmaxf(t, 0.0f);
    }
    float4 o = {hh[0], hh[1], hh[2], hh[3]};
    *(float4*)(h + (long)idx * 4) = o;
}

// ---------------------------------------------------------------------------
// Per-graph segment sums + counts. One wave per node.
__global__ __launch_bounds__(256) void readout_acc(
    const float* __restrict__ h, const int* __restrict__ gid,
    float* __restrict__ gsum, float* __restrict__ gcnt, int N)
{
    int node = (blockIdx.x * 256 + threadIdx.x) >> 5;
    int lane = threadIdx.x & 31;
    if (node >= N) return;
    int g = gid[node];
    float4 v = *(const float4*)(h + (long)node * HID + lane * 4);
    float* p = gsum + g * HID + lane * 4;
    atomAddF32(p + 0, v.x);
    atomAddF32(p + 1, v.y);
    atomAddF32(p + 2, v.z);
    atomAddF32(p + 3, v.w);
    if (lane == 0) atomAddF32(&gcnt[g], 1.0f);
}

// ---------------------------------------------------------------------------
// out[g,c] = (gsum[g]/max(cnt,1)) . ro_W[c] + ro_b[c]
__global__ void readout_final(const float* __restrict__ gsum, const float* __restrict__ gcnt,
                              const float* __restrict__ roW, const float* __restrict__ rob,
                              float* __restrict__ out)
{
    int idx = blockIdx.x * blockDim.x + threadIdx.x;
    if (idx >= N_GRAPHS * N_CLASSES) return;
    int g = idx / N_CLASSES, c = idx % N_CLASSES;
    float inv = 1.0f / fmaxf(gcnt[g], 1.0f);
    float s = 0.0f;
    #pragma unroll 8
    for (int k = 0; k < HID; ++k) s += gsum[g * HID + k] * inv * roW[c * HID + k];
    out[idx] = s + rob[c];
}

// ---------------------------------------------------------------------------
extern "C" void kernel_launch(void* const* d_in, const int* in_sizes, int n_in,
                              void* d_out, int out_size, void* d_ws, size_t ws_size,
                              hipStream_t stream) {
    const float* h_in   = (const float*)d_in[0];
    const float* snorm  = (const float*)d_in[1];
    const int*   src    = (const int*)  d_in[2];
    const int*   dst    = (const int*)  d_in[3];
    const int*   gid    = (const int*)  d_in[4];
    const float* eW     = (const float*)d_in[5];
    const float* eM     = (const float*)d_in[6];
    const float* eB     = (const float*)d_in[7];
    const float* lW     = (const float*)d_in[8];
    const float* lM     = (const float*)d_in[9];
    const float* lB     = (const float*)d_in[10];
    const float* gamma  = (const float*)d_in[11];
    const float* beta   = (const float*)d_in[12];
    const float* roW    = (const float*)d_in[13];
    const float* roB    = (const float*)d_in[14];

    const long NF = (long)N_NODES * HID;           // 6,400,000
    float* ws     = (float*)d_ws;
    float* h      = ws;
    float* agg    = h    + NF;
    float* z      = agg  + NF;
    float* wm     = z    + NF;                     // 5 * 16384
    float* bn_sum = wm   + (1 + N_LAYERS) * HID * HID;
    float* bn_sq  = bn_sum + HID;
    float* gsum   = bn_sq  + HID;                  // 128*128
    float* gcnt   = gsum + N_GRAPHS * HID;         // 128

    const int gemm_blocks = (N_NODES + 127) / 128;           // 391
    const int edge_blocks = (N_EDGES * 32 + 255) / 256;      // 75000
    const int node_blocks = (N_NODES * 32 + 255) / 256;

    // Precompute sparsity-masked weights
    mask_weights<<<((1 + N_LAYERS) * HID * HID + 255) / 256, 256, 0, stream>>>(eW, eM, lW, lM, wm);

    // Embedding: h = h_in @ Wm_e^T + b_e
    gemm_wmma<<<gemm_blocks, 256, 0, stream>>>(h_in, wm, eB, nullptr, h,
                                               nullptr, nullptr, N_NODES, 0);

    for (int l = 0; l < N_LAYERS; ++l) {
        zero_f32v4<<<(int)((NF / 4 + 255) / 256), 256, 0, stream>>>((float4*)agg, NF / 4);
        zero_f32v4<<<1, 64, 0, stream>>>((float4*)bn_sum, 2 * HID / 4);
        edge_agg<<<edge_blocks, 256, 0, stream>>>(h, src, dst, agg, N_EDGES);
        gemm_wmma<<<gemm_blocks, 256, 0, stream>>>(agg, wm + (1 + l) * HID * HID,
                                                   lB + l * HID, snorm, z,
                                                   bn_sum, bn_sq, N_NODES, 1);
        bn_apply<<<node_blocks, 256, 0, stream>>>(z, bn_sum, bn_sq,
                                                  gamma + l * HID, beta + l * HID,
                                                  h, N_NODES);
    }

    zero_f32v4<<<((N_GRAPHS * HID + N_GRAPHS) / 4 + 255) / 256, 256, 0, stream>>>(
        (float4*)gsum, (N_GRAPHS * HID + N_GRAPHS) / 4);
    readout_acc<<<node_blocks, 256, 0, stream>>>(h, gid, gsum, gcnt, N_NODES);
    readout_final<<<(N_GRAPHS * N_CLASSES + 255) / 256, 256, 0, stream>>>(gsum, gcnt, roW, roB, (float*)d_out);
}